// _PerturbPredictor_59811714564726
// MI455X (gfx1250) — compile-verified
//
#include <hip/hip_runtime.h>

// ---------------- problem constants (match reference) ----------------
constexpr int N_NODES = 100000;
constexpr int N_EDGES = 20000;   // 20000/16 = 1250 row tiles, exact
constexpr int N_PAIRS = 1600000;
constexpr int DIM     = 128;     // IN_DIM == HID_DIM

typedef float v2f __attribute__((ext_vector_type(2)));
typedef float v8f __attribute__((ext_vector_type(8)));

static inline size_t align256(size_t x) { return (x + 255) & ~size_t(255); }

// ---------------- k1: integer counts per node / per edge ----------------
__global__ __launch_bounds__(256) void k_count(const int* __restrict__ nidx,
                                               const int* __restrict__ eidx,
                                               int* __restrict__ cnt_n,
                                               int* __restrict__ cnt_e) {
    int p = blockIdx.x * 256 + threadIdx.x;
    if (p >= N_PAIRS) return;
    atomicAdd(cnt_n + nidx[p], 1);
    atomicAdd(cnt_e + eidx[p], 1);
}

// ---------------- k2: single-block chunked exclusive scan (Hillis-Steele in LDS) ----
__global__ __launch_bounds__(1024) void k_scan(const int* __restrict__ counts,
                                               int* __restrict__ offsets, int n) {
    __shared__ int buf[1024];
    __shared__ int carry_s;
    const int tid = threadIdx.x;
    if (tid == 0) carry_s = 0;
    for (int base = 0; base < n; base += 1024) {
        int i = base + tid;
        int v = (i < n) ? counts[i] : 0;
        __syncthreads();            // protect buf/carry from previous iteration
        buf[tid] = v;
        __syncthreads();
        for (int off = 1; off < 1024; off <<= 1) {
            int t = (tid >= off) ? buf[tid - off] : 0;
            __syncthreads();
            buf[tid] += t;
            __syncthreads();
        }
        int excl = carry_s + buf[tid] - v;   // exclusive prefix
        if (i < n) offsets[i] = excl;
        int total = buf[1023];
        __syncthreads();
        if (tid == 0) carry_s += total;
    }
}

// ---------------- k3: bin pairs into CSR lists (store payload ids directly) --------
__global__ __launch_bounds__(256) void k_bin(const int* __restrict__ nidx,
                                             const int* __restrict__ eidx,
                                             const int* __restrict__ off_e,
                                             const int* __restrict__ off_n,
                                             int* __restrict__ cur_e,
                                             int* __restrict__ cur_n,
                                             int* __restrict__ edge_nodes,
                                             int* __restrict__ node_edges) {
    int p = blockIdx.x * 256 + threadIdx.x;
    if (p >= N_PAIRS) return;
    int node = nidx[p];
    int e    = eidx[p];
    int pe = off_e[e]    + atomicAdd(cur_e + e, 1);
    edge_nodes[pe] = node;                 // member node id, CSR by edge
    int pn = off_n[node] + atomicAdd(cur_n + node, 1);
    node_edges[pn] = e;                    // incident edge id, CSR by node
}

// ---------------- k4: per-edge segment sum of x rows (no atomics) ------------------
// one wave per edge. 32 member ids are loaded coalesced (one per lane) and then
// broadcast with shfl, so the 512B row gathers have no index-load dependency and
// can be kept in flight back-to-back. Degree sum is lane-parallel.
__global__ __launch_bounds__(256) void k_edge_sum(const float* __restrict__ x,
                                                  const int* __restrict__ edge_nodes,
                                                  const int* __restrict__ off_e,
                                                  const int* __restrict__ cnt_e,
                                                  const int* __restrict__ cnt_n,
                                                  float* __restrict__ he_sum,
                                                  float* __restrict__ scale_e) {
    int e = blockIdx.x * 8 + (threadIdx.x >> 5);
    if (e >= N_EDGES) return;              // wave-uniform exit
    int lane  = threadIdx.x & 31;
    int start = off_e[e];
    int cnt   = cnt_e[e];
    float ax = 0.f, ay = 0.f, az = 0.f, aw = 0.f;
    float dsum = 0.f;
    for (int j0 = 0; j0 < cnt; j0 += 32) {
        int rem  = cnt - j0;
        int m    = rem < 32 ? rem : 32;
        int myid = (lane < m) ? edge_nodes[start + j0 + lane] : 0;   // coalesced
        if (lane < m) dsum += fmaxf((float)cnt_n[myid], 1.0f);        // clamped deg_n
        for (int jj = 0; jj < m; ++jj) {
            int node = __shfl(myid, jj, 32);                          // broadcast id
            float4 v = ((const float4*)(x + (size_t)node * DIM))[lane];
            ax += v.x; ay += v.y; az += v.z; aw += v.w;
        }
    }
#pragma unroll
    for (int msk = 16; msk >= 1; msk >>= 1)
        dsum += __shfl_xor(dsum, msk, 32);
    float4 o; o.x = ax; o.y = ay; o.z = az; o.w = aw;
    ((float4*)(he_sum + (size_t)e * DIM))[lane] = o;
    if (lane == 0) {
        float c     = fmaxf((float)cnt, 1.0f);
        float deg_e = dsum / c;                  // mean member-node degree
        scale_e[e]  = rsqrtf(deg_e) / c;         // folds (1/cnt) * 1/sqrt(deg_e)
    }
}

// ---------------- k5: m_e = (he_sum @ conv_w) * scale_e[row]  via V_WMMA_F32_16X16X4_F32
// block = 256 threads = 8 waves; wave w owns output tile [rowbase:+16) x [16w:+16)
__global__ __launch_bounds__(256) void k_gemm_wmma(const float* __restrict__ A,   // he_sum [E x 128]
                                                   const float* __restrict__ B,   // conv_w [128 x 128]
                                                   const float* __restrict__ scale_e,
                                                   float* __restrict__ C) {       // m_e [E x 128]
    const int lane    = threadIdx.x & 31;
    const int wave    = threadIdx.x >> 5;
    const int rowbase = blockIdx.x * 16;
    const int colbase = wave * 16;
    const int m       = lane & 15;          // A row / B col within tile
    const int khalf   = (lane >> 4) << 1;   // lanes 16..31 handle K+2,K+3

    const float* Arow = A + (size_t)(rowbase + m) * DIM;
    const int    col  = colbase + m;

    v8f c = {};
    for (int k0 = 0; k0 < DIM; k0 += 4) {
        v2f a;   // A 16x4: VGPR0 = K / K+2, VGPR1 = K+1 / K+3
        a.x = Arow[k0 + khalf + 0];
        a.y = Arow[k0 + khalf + 1];
        v2f b;   // B 4x16 mirrors A with N striped across lanes
        b.x = B[(size_t)(k0 + khalf + 0) * DIM + col];
        b.y = B[(size_t)(k0 + khalf + 1) * DIM + col];
        c = __builtin_amdgcn_wmma_f32_16x16x4_f32(false, a, false, b,
                                                  (short)0, c, false, false);
    }

    const int rofs = (lane >> 4) * 8;       // C: VGPR r -> row r or r+8
#pragma unroll
    for (int r = 0; r < 8; ++r) {
        int row = rowbase + rofs + r;
        C[(size_t)row * DIM + col] = c[r] * scale_e[row];
    }
}

// ---------------- k6: per-node gather of m_e rows + fused normalize/ReLU/readout ----
// same shfl-broadcast gather structure as k_edge_sum
__global__ __launch_bounds__(256) void k_node_out(const float* __restrict__ m_e,
                                                  const int* __restrict__ node_edges,
                                                  const int* __restrict__ off_n,
                                                  const int* __restrict__ cnt_n,
                                                  const float* __restrict__ conv_b,
                                                  const float* __restrict__ read_w,
                                                  const float* __restrict__ read_b,
                                                  float* __restrict__ out) {
    int node = blockIdx.x * 8 + (threadIdx.x >> 5);
    if (node >= N_NODES) return;           // wave-uniform exit
    int lane  = threadIdx.x & 31;
    int start = off_n[node];
    int cnt   = cnt_n[node];
    float ax = 0.f, ay = 0.f, az = 0.f, aw = 0.f;
    for (int j0 = 0; j0 < cnt; j0 += 32) {
        int rem = cnt - j0;
        int m   = rem < 32 ? rem : 32;
        int mye = (lane < m) ? node_edges[start + j0 + lane] : 0;    // coalesced
        for (int jj = 0; jj < m; ++jj) {
            int e = __shfl(mye, jj, 32);                             // broadcast id
            float4 v = ((const float4*)(m_e + (size_t)e * DIM))[lane];
            ax += v.x; ay += v.y; az += v.z; aw += v.w;
        }
    }
    float inv = rsqrtf(fmaxf((float)cnt, 1.0f));    // 1/sqrt(deg_n), clamped
    float4 b = ((const float4*)conv_b)[lane];
    float4 w = ((const float4*)read_w)[lane];
    float h0 = fmaxf(fmaf(ax, inv, b.x), 0.0f);
    float h1 = fmaxf(fmaf(ay, inv, b.y), 0.0f);
    float h2 = fmaxf(fmaf(az, inv, b.z), 0.0f);
    float h3 = fmaxf(fmaf(aw, inv, b.w), 0.0f);
    float dot = h0 * w.x + h1 * w.y + h2 * w.z + h3 * w.w;
#pragma unroll
    for (int msk = 16; msk >= 1; msk >>= 1)
        dot += __shfl_xor(dot, msk, 32);
    if (lane == 0) out[node] = dot + read_b[0];
}

// ---------------- host-side launcher ----------------
extern "C" void kernel_launch(void* const* d_in, const int* in_sizes, int n_in,
                              void* d_out, int out_size, void* d_ws, size_t ws_size,
                              hipStream_t stream) {
    const float* x      = (const float*)d_in[0];
    const float* conv_w = (const float*)d_in[1];
    const float* conv_b = (const float*)d_in[2];
    const float* read_w = (const float*)d_in[3];
    const float* read_b = (const float*)d_in[4];
    const int*   nidx   = (const int*)d_in[5];
    const int*   eidx   = (const int*)d_in[6];
    float*       out    = (float*)d_out;

    // workspace layout: zero-init region first (counts + cursors), then the rest
    char* w = (char*)d_ws;
    size_t o_cnt_n  = 0;
    size_t o_cnt_e  = o_cnt_n  + align256((size_t)N_NODES * 4);
    size_t o_cur_n  = o_cnt_e  + align256((size_t)N_EDGES * 4);
    size_t o_cur_e  = o_cur_n  + align256((size_t)N_NODES * 4);
    size_t o_zend   = o_cur_e  + align256((size_t)N_EDGES * 4);   // end of zeroed region
    size_t o_off_n  = o_zend;
    size_t o_off_e  = o_off_n  + align256((size_t)N_NODES * 4);
    size_t o_scale  = o_off_e  + align256((size_t)N_EDGES * 4);
    size_t o_enod   = o_scale  + align256((size_t)N_EDGES * 4);
    size_t o_nedg   = o_enod   + align256((size_t)N_PAIRS * 4);
    size_t o_he     = o_nedg   + align256((size_t)N_PAIRS * 4);
    size_t o_me     = o_he     + align256((size_t)N_EDGES * DIM * 4);

    int*   cnt_n      = (int*)(w + o_cnt_n);
    int*   cnt_e      = (int*)(w + o_cnt_e);
    int*   cur_n      = (int*)(w + o_cur_n);
    int*   cur_e      = (int*)(w + o_cur_e);
    int*   off_n      = (int*)(w + o_off_n);
    int*   off_e      = (int*)(w + o_off_e);
    float* scale_e    = (float*)(w + o_scale);
    int*   edge_nodes = (int*)(w + o_enod);
    int*   node_edges = (int*)(w + o_nedg);
    float* he_sum     = (float*)(w + o_he);
    float* m_e        = (float*)(w + o_me);

    hipMemsetAsync(w, 0, o_zend, stream);

    k_count<<<(N_PAIRS + 255) / 256, 256, 0, stream>>>(nidx, eidx, cnt_n, cnt_e);

    k_scan<<<1, 1024, 0, stream>>>(cnt_e, off_e, N_EDGES);
    k_scan<<<1, 1024, 0, stream>>>(cnt_n, off_n, N_NODES);

    k_bin<<<(N_PAIRS + 255) / 256, 256, 0, stream>>>(nidx, eidx, off_e, off_n,
                                                     cur_e, cur_n, edge_nodes, node_edges);

    k_edge_sum<<<(N_EDGES + 7) / 8, 256, 0, stream>>>(x, edge_nodes, off_e, cnt_e,
                                                      cnt_n, he_sum, scale_e);

    k_gemm_wmma<<<N_EDGES / 16, 256, 0, stream>>>(he_sum, conv_w, scale_e, m_e);

    k_node_out<<<(N_NODES + 7) / 8, 256, 0, stream>>>(m_e, node_edges, off_n, cnt_n,
                                                      conv_b, read_w, read_b, out);
}